// CancerDetectionValidRegionLoss_15908558864773
// MI455X (gfx1250) — compile-verified
//
#include <hip/hip_runtime.h>
#include <hip/hip_bf16.h>
#include <stdint.h>

// Masked BCE-with-logits mean over valid region.
//   mask = (prostate>0.5)&(needle>0.5)
//   bce  = max(x,0) - x*y + log1p(exp(-|x|))
//   out  = sum(bce*mask) / max(sum(mask), 1)
//
// Memory-bound (~100.7 MB read, 1 float out) -> HBM roofline ~4.3us @ 23.3TB/s.
// Data path: double-buffered GLOBAL_LOAD_ASYNC_TO_LDS_B128 (ASYNCcnt) pipeline,
// hardware transcendentals (v_exp_f32 / v_log_f32) to keep VALU off the
// critical path, per-wave shfl reduction, deterministic two-pass final reduce.

#define TPB 256
#define WAVES_PER_BLOCK (TPB / 32)
#define GRIDX 64

__device__ __forceinline__ unsigned lds_offset_of(const void* p) {
    // Flat addr of an LDS location = {aperture_hi32, lds_offset32}; low 32 bits
    // are the wave-relative LDS byte address used by DS/async instructions.
    return (unsigned)(uintptr_t)p;
}

__device__ __forceinline__ void async_load_b128(unsigned lds_addr, unsigned voff,
                                                unsigned long long gbase) {
    asm volatile("global_load_async_to_lds_b128 %0, %1, %2"
                 :
                 : "v"(lds_addr), "v"(voff), "s"(gbase)
                 : "memory");
}

__device__ __forceinline__ float bce_term(float x, float y) {
    // softplus(-|x|) via hardware exp/log; abs error <= ~6e-8 for |x|>16.6
    float t = __expf(-fabsf(x));
    return fmaxf(x, 0.0f) - x * y + __logf(1.0f + t);
}

__global__ __launch_bounds__(TPB) void bce_pass1(
    const float* __restrict__ logits, const float* __restrict__ label,
    const float* __restrict__ pmask, const float* __restrict__ nmask,
    float* __restrict__ part, int hw, int nvec, int vpb, int rem, int nPart) {
    __shared__ __align__(16) float lds_x[2][TPB * 4];
    __shared__ __align__(16) float lds_p[2][TPB * 4];
    __shared__ __align__(16) float lds_n[2][TPB * 4];
    __shared__ float red[2 * WAVES_PER_BLOCK];

    const int tid = threadIdx.x;
    const int b = blockIdx.y;
    const long long imgOff = (long long)b * (long long)hw;  // element offset
    const float y = label[b];

    const unsigned long long gx = (unsigned long long)(logits + imgOff);
    const unsigned long long gp = (unsigned long long)(pmask + imgOff);
    const unsigned long long gn = (unsigned long long)(nmask + imgOff);

    const int start = blockIdx.x * vpb;
    const int end = min(start + vpb, nvec);
    const int ntile = (end > start) ? ((end - start + TPB - 1) / TPB) : 0;

    const unsigned lx0 = lds_offset_of(&lds_x[0][tid * 4]);
    const unsigned lp0 = lds_offset_of(&lds_p[0][tid * 4]);
    const unsigned ln0 = lds_offset_of(&lds_n[0][tid * 4]);
    const unsigned lx1 = lds_offset_of(&lds_x[1][tid * 4]);
    const unsigned lp1 = lds_offset_of(&lds_p[1][tid * 4]);
    const unsigned ln1 = lds_offset_of(&lds_n[1][tid * 4]);

    // Issue tile t into buffer (t & 1). Lanes past the chunk end are clamped
    // (full-EXEC issue) and masked out at consume time.
    auto issue = [&](int t) {
        int v = start + t * TPB + tid;
        v = (v < end) ? v : (end - 1);
        unsigned off = (unsigned)v * 16u;  // byte offset (float4)
        if (t & 1) {
            async_load_b128(lx1, off, gx);
            async_load_b128(lp1, off, gp);
            async_load_b128(ln1, off, gn);
        } else {
            async_load_b128(lx0, off, gx);
            async_load_b128(lp0, off, gp);
            async_load_b128(ln0, off, gn);
        }
    };

    float accS = 0.0f, accC = 0.0f;

    if (ntile > 0) {
        issue(0);
        for (int t = 0; t < ntile; ++t) {
            if (t + 1 < ntile) {
                issue(t + 1);
                // oldest tile (3 async loads) complete; next tile still in flight
                asm volatile("s_wait_asynccnt 0x3" ::: "memory");
            } else {
                asm volatile("s_wait_asynccnt 0x0" ::: "memory");
            }
            const int bufi = t & 1;
            const int v = start + t * TPB + tid;
            const bool valid = (v < end);
#pragma unroll
            for (int c = 0; c < 4; ++c) {
                float x = lds_x[bufi][tid * 4 + c];
                float p = lds_p[bufi][tid * 4 + c];
                float n = lds_n[bufi][tid * 4 + c];
                float m = (valid && p > 0.5f && n > 0.5f) ? 1.0f : 0.0f;
                accS += bce_term(x, y) * m;
                accC += m;
            }
        }
    }

    // Scalar tail (hw % 4), handled once per image by block x==0.
    if (blockIdx.x == 0 && tid < rem) {
        long long i = imgOff + (long long)nvec * 4 + tid;
        float x = logits[i], p = pmask[i], n = nmask[i];
        float m = (p > 0.5f && n > 0.5f) ? 1.0f : 0.0f;
        accS += bce_term(x, y) * m;
        accC += m;
    }

    // wave32 reduction
#pragma unroll
    for (int o = 16; o > 0; o >>= 1) {
        accS += __shfl_xor(accS, o, 32);
        accC += __shfl_xor(accC, o, 32);
    }
    const int wave = tid >> 5, lane = tid & 31;
    if (lane == 0) {
        red[wave] = accS;
        red[WAVES_PER_BLOCK + wave] = accC;
    }
    __syncthreads();
    if (tid == 0) {
        float s = 0.0f, c = 0.0f;
#pragma unroll
        for (int w = 0; w < WAVES_PER_BLOCK; ++w) {
            s += red[w];
            c += red[WAVES_PER_BLOCK + w];
        }
        const int pidx = b * gridDim.x + blockIdx.x;
        part[pidx] = s;
        part[nPart + pidx] = c;
    }
}

__global__ __launch_bounds__(TPB) void bce_pass2(const float* __restrict__ part,
                                                 int nPart,
                                                 float* __restrict__ out) {
    __shared__ float rs[WAVES_PER_BLOCK], rc[WAVES_PER_BLOCK];
    float s = 0.0f, c = 0.0f;
    for (int i = threadIdx.x; i < nPart; i += TPB) {
        s += part[i];
        c += part[nPart + i];
    }
#pragma unroll
    for (int o = 16; o > 0; o >>= 1) {
        s += __shfl_xor(s, o, 32);
        c += __shfl_xor(c, o, 32);
    }
    const int wave = threadIdx.x >> 5, lane = threadIdx.x & 31;
    if (lane == 0) {
        rs[wave] = s;
        rc[wave] = c;
    }
    __syncthreads();
    if (threadIdx.x == 0) {
        float S = 0.0f, C = 0.0f;
#pragma unroll
        for (int w = 0; w < WAVES_PER_BLOCK; ++w) {
            S += rs[w];
            C += rc[w];
        }
        out[0] = S / fmaxf(C, 1.0f);
    }
}

extern "C" void kernel_launch(void* const* d_in, const int* in_sizes, int n_in,
                              void* d_out, int out_size, void* d_ws, size_t ws_size,
                              hipStream_t stream) {
    const float* logits = (const float*)d_in[0];   // (B,1,H,W) f32
    const float* label  = (const float*)d_in[1];   // (B,)      f32
    const float* pmask  = (const float*)d_in[2];   // (B,1,H,W) f32
    const float* nmask  = (const float*)d_in[3];   // (B,1,H,W) f32

    const int N = in_sizes[0];
    const int B = in_sizes[1];
    const int hw = N / B;          // 262144
    const int nvec = hw / 4;       // float4 count per image
    const int rem = hw - nvec * 4;

    int gridx = GRIDX;
    // Ensure partials (2 * gridx * B floats) fit in workspace.
    while (gridx > 1 && (size_t)(2ull * gridx * B * sizeof(float)) > ws_size)
        gridx >>= 1;
    const int vpb = (nvec + gridx - 1) / gridx;
    const int nPart = gridx * B;

    float* part = (float*)d_ws;
    dim3 grid(gridx, B);
    bce_pass1<<<grid, TPB, 0, stream>>>(logits, label, pmask, nmask, part,
                                        hw, nvec, vpb, rem, nPart);
    bce_pass2<<<1, TPB, 0, stream>>>(part, nPart, (float*)d_out);
}